// MPCACTLayer_87230785782541
// MI455X (gfx1250) — compile-verified
//
#include <hip/hip_runtime.h>
#include <hip/hip_bf16.h>

typedef float v2f __attribute__((ext_vector_type(2)));
typedef float v8f __attribute__((ext_vector_type(8)));

#define OBS_DIM 128
#define HID     256
#define XD      22
#define AD      4
#define QCOLS   52      // (XD+AD)*2
#define HORIZON 10
#define DTC     0.1f
#define EPSC    1e-4f
#define NIT     8
#define LRC     0.1f
#define HS      260     // LDS row stride for h (pad to kill bank conflicts)

__constant__ int c_idx[XD] = {39,40,41,42,43,44,45,46,47,48,
                              0,1,2, 13,14,15, 26,27,28, 69,70,71};

__device__ __forceinline__ float softplus_f(float x) {
    // logaddexp(x, 0) = max(x,0) + log1p(exp(-|x|))   (matches jax.nn.softplus)
    return fmaxf(x, 0.0f) + log1pf(expf(-fabsf(x)));
}
__device__ __forceinline__ float mish_f(float x) {
    return x * tanhf(softplus_f(x));
}
__device__ __forceinline__ v8f v8f_zero() {
    v8f z;
#pragma unroll
    for (int j = 0; j < 8; ++j) z[j] = 0.0f;
    return z;
}

// ---------------------------------------------------------------------------
// Kernel 1: h = mish(obs@W1 + b1) [staged in LDS]; q = softplus(h@W2+b2)+eps
// block = 128 threads (4 waves), each wave owns a 16-row strip; 64 rows/block.
// f32 WMMA 16x16x4 for both GEMMs. A-fragments register-cached; per-tile
// epilogue keeps accumulator pressure at 16 VGPRs (no scratch spill).
// ---------------------------------------------------------------------------
__global__ __launch_bounds__(128)
void fused_mlp_kernel(const float* __restrict__ obs,
                      const float* __restrict__ W1, const float* __restrict__ bias1,
                      const float* __restrict__ W2, const float* __restrict__ bias2,
                      float* __restrict__ qout)
{
    extern __shared__ float hbuf[];              // 64 x HS floats
    const int lane = threadIdx.x & 31;
    const int wave = threadIdx.x >> 5;
    const int m    = lane & 15;                  // row-in-tile (A) / col (B/C)
    const int koff = (lane >> 4) * 2;            // 0 or 2: K sub-slot
    const int rhalf = (lane >> 4) * 8;           // C/D row offset for hi lanes
    const int rowBase = blockIdx.x * 64 + wave * 16;

    // ---- GEMM1: 16x256 strip, K = 128 -------------------------------------
    // Preload all A-fragments (obs read exactly once): 32 x v2f = 64 VGPRs.
    v2f afrag[32];
    const float* aRow = obs + (size_t)(rowBase + m) * OBS_DIM;
#pragma unroll
    for (int kk = 0; kk < 32; ++kk) {
        const int kb = kk * 4 + koff;
        afrag[kk].x = aRow[kb];
        afrag[kk].y = aRow[kb + 1];
    }

    // Two N-tiles per iteration (independent accumulators break the serial
    // WMMA->WMMA dependency); mish epilogue runs per pair -> low reg pressure.
    for (int nt = 0; nt < 16; nt += 2) {
        const int col0 = nt * 16 + m;
        const int col1 = col0 + 16;
        v8f acc0 = v8f_zero();
        v8f acc1 = v8f_zero();
#pragma unroll
        for (int kk = 0; kk < 32; ++kk) {
            const int kb = kk * 4 + koff;
            v2f b0, b1f;
            b0.x  = W1[kb * HID + col0];
            b0.y  = W1[(kb + 1) * HID + col0];
            b1f.x = W1[kb * HID + col1];
            b1f.y = W1[(kb + 1) * HID + col1];
            acc0 = __builtin_amdgcn_wmma_f32_16x16x4_f32(
                false, afrag[kk], false, b0, (short)0, acc0, false, false);
            acc1 = __builtin_amdgcn_wmma_f32_16x16x4_f32(
                false, afrag[kk], false, b1f, (short)0, acc1, false, false);
        }
        const float bv0 = bias1[col0];
        const float bv1 = bias1[col1];
#pragma unroll
        for (int j = 0; j < 8; ++j) {
            const int r = (wave * 16 + rhalf + j) * HS;
            hbuf[r + col0] = mish_f(acc0[j] + bv0);
            hbuf[r + col1] = mish_f(acc1[j] + bv1);
        }
    }
    __syncthreads();

    // ---- GEMM2: 16x64(pad of 52) tile, K = 256 from LDS -------------------
    v8f acc2[4];
#pragma unroll
    for (int nt = 0; nt < 4; ++nt) acc2[nt] = v8f_zero();

    const float* hRow = hbuf + (size_t)(wave * 16 + m) * HS;
    for (int kk = 0; kk < 64; ++kk) {
        const int kb = kk * 4 + koff;
        v2f a;
        a.x = hRow[kb];
        a.y = hRow[kb + 1];
#pragma unroll
        for (int nt = 0; nt < 4; ++nt) {
            const int col = nt * 16 + m;
            const int cc  = col < QCOLS ? col : (QCOLS - 1);  // clamp (in-bounds)
            v2f b;
            float bx = W2[kb * QCOLS + cc];
            float by = W2[(kb + 1) * QCOLS + cc];
            b.x = (col < QCOLS) ? bx : 0.0f;                  // mask pad cols
            b.y = (col < QCOLS) ? by : 0.0f;
            acc2[nt] = __builtin_amdgcn_wmma_f32_16x16x4_f32(
                false, a, false, b, (short)0, acc2[nt], false, false);
        }
    }

#pragma unroll
    for (int nt = 0; nt < 4; ++nt) {
        const int col = nt * 16 + m;
        if (col < QCOLS) {
            const float bv = bias2[col];
#pragma unroll
            for (int j = 0; j < 8; ++j) {
                const int row = rowBase + rhalf + j;
                qout[(size_t)row * QCOLS + col] =
                    softplus_f(acc2[nt][j] + bv) + EPSC;
            }
        }
    }
}

// ---------------------------------------------------------------------------
// Kernel 2: per-row MPC solve, 8 GD iterations with adjoint gradient.
// One thread per batch row; A/Bm broadcast from LDS; x-history in LDS.
// ---------------------------------------------------------------------------
__global__ __launch_bounds__(64)
void mpc_kernel(const float* __restrict__ obs, const float* __restrict__ qbuf,
                const float* __restrict__ Amat, const float* __restrict__ Bmat,
                const float* __restrict__ actor_std,
                float* __restrict__ out, int B)
{
    __shared__ float sA[XD * XD];
    __shared__ float sB[XD * AD];
    __shared__ float xs[HORIZON * XD * 64];      // [t*XD+i][tid] layout

    for (int i = threadIdx.x; i < XD * XD; i += blockDim.x) sA[i] = Amat[i];
    for (int i = threadIdx.x; i < XD * AD; i += blockDim.x) sB[i] = Bmat[i];
    __syncthreads();

    const int tid = threadIdx.x;
    const int row = blockIdx.x * 64 + tid;

    float x0[XD], Q[XD], Qf[XD], R[AD], Rf[AD];
    const float* orow = obs + (size_t)row * OBS_DIM;
#pragma unroll
    for (int i = 0; i < XD; ++i) x0[i] = orow[c_idx[i]];
    const float* qrow = qbuf + (size_t)row * QCOLS;
#pragma unroll
    for (int i = 0; i < XD; ++i) Q[i] = qrow[i];
#pragma unroll
    for (int a = 0; a < AD; ++a) R[a] = qrow[XD + a];
#pragma unroll
    for (int i = 0; i < XD; ++i) Qf[i] = qrow[XD + AD + i];
#pragma unroll
    for (int a = 0; a < AD; ++a) Rf[a] = qrow[2 * XD + AD + a];

    float U[HORIZON][AD];
#pragma unroll
    for (int t = 0; t < HORIZON; ++t)
#pragma unroll
        for (int a = 0; a < AD; ++a) U[t][a] = 0.0f;

    for (int it = 0; it < NIT; ++it) {
        // forward rollout, stash x_t
        float x[XD];
#pragma unroll
        for (int i = 0; i < XD; ++i) x[i] = x0[i];
#pragma unroll
        for (int t = 0; t < HORIZON; ++t) {
#pragma unroll
            for (int i = 0; i < XD; ++i) xs[(t * XD + i) * 64 + tid] = x[i];
            float xn[XD];
#pragma unroll
            for (int i = 0; i < XD; ++i) {
                float s = 0.0f;
#pragma unroll
                for (int j = 0; j < XD; ++j) s += sA[i * XD + j] * x[j];
#pragma unroll
                for (int a = 0; a < AD; ++a) s += sB[i * AD + a] * U[t][a];
                xn[i] = x[i] + DTC * s;
            }
#pragma unroll
            for (int i = 0; i < XD; ++i) x[i] = xn[i];
        }

        // adjoint backward, update U in place
        float lam[XD];
#pragma unroll
        for (int i = 0; i < XD; ++i) lam[i] = 2.0f * Qf[i] * x[i];   // lambda_H
#pragma unroll
        for (int t = HORIZON - 1; t >= 0; --t) {
            float g[AD];
#pragma unroll
            for (int a = 0; a < AD; ++a) {
                float s = 0.0f;
#pragma unroll
                for (int i = 0; i < XD; ++i) s += sB[i * AD + a] * lam[i];
                g[a] = 2.0f * R[a] * U[t][a] + DTC * s;
            }
            if (t == HORIZON - 1) {
#pragma unroll
                for (int a = 0; a < AD; ++a) g[a] += 2.0f * Rf[a] * U[t][a];
            }
#pragma unroll
            for (int a = 0; a < AD; ++a) U[t][a] -= LRC * g[a];
            // lambda_t = 2 Q∘x_t + lam + DT * A^T lam
            float nl[XD];
#pragma unroll
            for (int i = 0; i < XD; ++i) {
                float s = 0.0f;
#pragma unroll
                for (int j = 0; j < XD; ++j) s += sA[j * XD + i] * lam[j];
                nl[i] = 2.0f * Q[i] * xs[(t * XD + i) * 64 + tid] + lam[i] + DTC * s;
            }
#pragma unroll
            for (int i = 0; i < XD; ++i) lam[i] = nl[i];
        }
    }

    // outputs: loc = U[0], scale = exp(actor_std)
#pragma unroll
    for (int a = 0; a < AD; ++a) {
        out[(size_t)row * AD + a] = U[0][a];
        out[(size_t)B * AD + (size_t)row * AD + a] = expf(actor_std[a]);
    }
}

// ---------------------------------------------------------------------------
extern "C" void kernel_launch(void* const* d_in, const int* in_sizes, int n_in,
                              void* d_out, int out_size, void* d_ws, size_t ws_size,
                              hipStream_t stream) {
    const float* obs       = (const float*)d_in[0];
    const float* W1        = (const float*)d_in[1];
    const float* b1v       = (const float*)d_in[2];
    const float* W2        = (const float*)d_in[3];
    const float* b2v       = (const float*)d_in[4];
    const float* actor_std = (const float*)d_in[5];
    const float* Amat      = (const float*)d_in[6];
    const float* Bmat      = (const float*)d_in[7];
    float* out  = (float*)d_out;
    float* qbuf = (float*)d_ws;                         // B x 52 floats

    const int B = in_sizes[0] / OBS_DIM;                // 65536

    const size_t lds1 = (size_t)64 * HS * sizeof(float); // 66,560 B dynamic LDS
    fused_mlp_kernel<<<dim3(B / 64), dim3(128), lds1, stream>>>(
        obs, W1, b1v, W2, b2v, qbuf);

    mpc_kernel<<<dim3(B / 64), dim3(64), 0, stream>>>(
        obs, qbuf, Amat, Bmat, actor_std, out, B);
}